// ToyPac_283467841829
// MI455X (gfx1250) — compile-verified
//
#include <hip/hip_runtime.h>
#include <hip/hip_bf16.h>

typedef float v2f __attribute__((ext_vector_type(2)));
typedef float v8f __attribute__((ext_vector_type(8)));

#define HW 65536          // 256*256
#define KTAPS 25
#define CIN 32
#define COUT 32
#define KLEN 800          // 25 taps * 32 channels
#define CPAD 34           // channel stride in LDS slab (bank-conflict pad, keeps 8B align)
#define SLABW 68          // 64-px segment + 4 halo columns

// ---------------------------------------------------------------------------
// Kernel 1: per-pixel PAC weights.
// kern[tap][p] = exp(-0.5 * sum_c (g_pad[c, y+i-2, x+j-2] - g[c,y,x])^2)
// im2col zero-pads: OOB neighbor value is literally 0 (d = -center).
// ---------------------------------------------------------------------------
__global__ __launch_bounds__(256) void pac_weights_kernel(
    const float* __restrict__ g, float* __restrict__ kout) {
  const int p = blockIdx.x * 256 + threadIdx.x;   // 0..65535
  const int y = p >> 8;
  const int x = p & 255;

  float gc[8];
#pragma unroll
  for (int c = 0; c < 8; ++c) gc[c] = g[c * HW + p];

#pragma unroll
  for (int tap = 0; tap < KTAPS; ++tap) {
    const int ti = tap / 5, tj = tap % 5;         // compile-time (unrolled)
    const int yy = y + ti - 2;
    const int xx = x + tj - 2;
    const bool inb = ((unsigned)yy < 256u) && ((unsigned)xx < 256u);
    const int sb = yy * 256 + xx;
    float s = 0.0f;
#pragma unroll
    for (int c = 0; c < 8; ++c) {
      const float gv = inb ? g[c * HW + sb] : 0.0f;
      const float d = gv - gc[c];
      s += d * d;
    }
    kout[tap * HW + p] = __expf(-0.5f * s);
  }
}

// ---------------------------------------------------------------------------
// Kernel 2: weight transpose  w[o][c][i][j]  ->  wt[o][tap*32 + c]
// (K reordered tap-major / channel-minor for the WMMA loop)
// ---------------------------------------------------------------------------
__global__ __launch_bounds__(256) void wtranspose_kernel(
    const float* __restrict__ w, float* __restrict__ wt) {
  const int t = blockIdx.x * 256 + threadIdx.x;
  if (t >= COUT * KLEN) return;
  const int o = t / KLEN;
  const int r = t - o * KLEN;
  const int tap = r >> 5;
  const int c = r & 31;
  wt[t] = w[o * KLEN + c * KTAPS + tap];
}

// ---------------------------------------------------------------------------
// Kernel 3: PAC conv + bias + ReLU via V_WMMA_F32_16X16X4_F32.
// Block = 128 threads (4 waves) = one 64-pixel segment of one row.
// LDS: zero-padded activation halo slab [5 rows][68 px][34 ch] (synchronous
// staging, zero-filled halo) + PAC-kernel slab (async copy engine:
// GLOBAL_LOAD_ASYNC_TO_LDS_B32, always in-bounds, ASYNCcnt-tracked).
// Each wave computes a 16-pixel x 32-oc tile:
//   D[oc][px] = sum_{tap,c} wt[oc][tap*32+c] * (x_pad[c,yy,xx] * kern[tap,y,px])
// A (16x4, oc x k):  lane m = lane&15, K pair = k0 + 2*(lane>>4) + {0,1}
// B (4x16, k x px):  lane n = lane&15, same K mapping -> one ds_load_b64
// D (16x16):         lane = px, VGPR v -> oc = v + 8*(lane>>4)
// ---------------------------------------------------------------------------
__global__ __launch_bounds__(128) void pac_conv_wmma_kernel(
    const float* __restrict__ src,    // [32][256][256]
    const float* __restrict__ wt,     // [32][800] (tap-major K)
    const float* __restrict__ bias,   // [32]
    const float* __restrict__ kern,   // [25][65536]
    float* __restrict__ dst) {        // [32][256][256]
  __shared__ float slab[5 * SLABW * CPAD];   // 46,240 B
  __shared__ float kslab[KTAPS * 64];        //  6,400 B

  const int seg = blockIdx.x;                // 1024 segments = 256 rows * 4
  const int y = seg >> 2;
  const int x0 = (seg & 3) << 6;

  // ---- async-stage PAC kernel values (all in-bounds) via the copy engine ----
  {
    const float* gbase = kern + y * 256 + x0;      // + tap*HW + pxl
    for (int t = threadIdx.x; t < KTAPS * 64; t += 128) {
      const int tap = t >> 6;
      const int pxl = t & 63;
      const unsigned lds_addr = (unsigned)(uintptr_t)(&kslab[t]);
      const float* gp = gbase + tap * HW + pxl;
      asm volatile("global_load_async_to_lds_b32 %0, %1, off"
                   :: "v"(lds_addr), "v"(gp)
                   : "memory");
    }
  }

  // ---- stage zero-padded activation slab: slab[(row*68+px)*34 + c] ----
  for (int t = threadIdx.x; t < CIN * 5 * SLABW; t += 128) {
    const int c = t / (5 * SLABW);
    const int r = t - c * (5 * SLABW);
    const int row = r / SLABW;
    const int pxl = r - row * SLABW;
    const int yy = y + row - 2;
    const int xx = x0 + pxl - 2;
    float v = 0.0f;
    if (((unsigned)yy < 256u) && ((unsigned)xx < 256u))
      v = src[c * HW + yy * 256 + xx];
    slab[(row * SLABW + pxl) * CPAD + c] = v;
  }

  // All of this wave's async LDS writes must land before the barrier.
  asm volatile("s_wait_asynccnt 0x0" ::: "memory");
  __syncthreads();

  const int lane = threadIdx.x & 31;
  const int wave = threadIdx.x >> 5;
  const int n = lane & 15;                  // A row (oc) / B col (pixel)
  const int khalf = (lane >> 4) << 1;       // 0 or 2: K offset inside 4-group
  const int pxl = (wave << 4) + n;          // pixel within segment, 0..63

  const float* wrow0 = wt + n * KLEN;            // oc = n
  const float* wrow1 = wt + (n + 16) * KLEN;     // oc = n + 16
  __builtin_prefetch(wrow0, 0, 3);               // global_prefetch_b8: warm WGP$
  __builtin_prefetch(wrow1, 0, 3);

  v8f acc0 = {};
  v8f acc1 = {};

  for (int tap = 0; tap < KTAPS; ++tap) {
    const int ti = tap / 5, tj = tap % 5;
    const float kv = kslab[(tap << 6) + pxl];              // PAC weight (tap,px)
    const int sb = (ti * SLABW + pxl + tj) * CPAD + khalf; // + c0 -> slab index
    const int kt = tap * 32 + khalf;                       // + c0 -> K index

#pragma unroll
    for (int c0 = 0; c0 < CIN; c0 += 4) {
      // B fragment: two adjacent channels from LDS (ds_load_b64), scaled
      v2f bf = *(const v2f*)&slab[sb + c0];
      bf.x *= kv;
      bf.y *= kv;
      // A fragments: transposed weights (8B-aligned global_load_b64, L2-hot)
      v2f a0 = *(const v2f*)(wrow0 + kt + c0);
      v2f a1 = *(const v2f*)(wrow1 + kt + c0);

      acc0 = __builtin_amdgcn_wmma_f32_16x16x4_f32(
          false, a0, false, bf, (short)0, acc0, false, false);
      acc1 = __builtin_amdgcn_wmma_f32_16x16x4_f32(
          false, a1, false, bf, (short)0, acc1, false, false);
    }
  }

  // Store: lane holds pixel; VGPR v holds oc = v + 8*(lane>>4) (+16 for acc1)
  const int mofs = (lane >> 4) << 3;       // 0 or 8
  const int pbase = y * 256 + x0 + pxl;
#pragma unroll
  for (int v = 0; v < 8; ++v) {
    const int oc0 = v + mofs;
    float r0 = acc0[v] + bias[oc0];
    dst[oc0 * HW + pbase] = r0 > 0.0f ? r0 : 0.0f;
    const int oc1 = oc0 + 16;
    float r1 = acc1[v] + bias[oc1];
    dst[oc1 * HW + pbase] = r1 > 0.0f ? r1 : 0.0f;
  }
}

// ---------------------------------------------------------------------------
// Launch: kern -> wt1/wt2 -> conv1(x->h) -> conv2(h->out)
// ---------------------------------------------------------------------------
extern "C" void kernel_launch(void* const* d_in, const int* in_sizes, int n_in,
                              void* d_out, int out_size, void* d_ws, size_t ws_size,
                              hipStream_t stream) {
  const float* x  = (const float*)d_in[0];   // (1,32,256,256)
  const float* g  = (const float*)d_in[1];   // (1,8,256,256)
  const float* w1 = (const float*)d_in[2];   // (32,32,5,5)
  const float* b1 = (const float*)d_in[3];   // (32,)
  const float* w2 = (const float*)d_in[4];   // (32,32,5,5)
  const float* b2 = (const float*)d_in[5];   // (32,)
  float* out = (float*)d_out;                // (1,32,256,256)

  float* ws   = (float*)d_ws;
  float* kern = ws;                          // 25*65536   = 1,638,400 f
  float* h    = kern + KTAPS * HW;           // 32*65536   = 2,097,152 f
  float* wt1  = h + COUT * HW;               // 32*800     =    25,600 f
  float* wt2  = wt1 + COUT * KLEN;           // 32*800     =    25,600 f

  pac_weights_kernel<<<256, 256, 0, stream>>>(g, kern);
  wtranspose_kernel<<<(COUT * KLEN + 255) / 256, 256, 0, stream>>>(w1, wt1);
  wtranspose_kernel<<<(COUT * KLEN + 255) / 256, 256, 0, stream>>>(w2, wt2);
  // 1024 blocks x 128 threads: 4 waves per 64-pixel row segment
  pac_conv_wmma_kernel<<<1024, 128, 0, stream>>>(x, wt1, b1, kern, h);
  pac_conv_wmma_kernel<<<1024, 128, 0, stream>>>(h, wt2, b2, kern, out);
}